// TFAdaptiveSoftmaxMask_49246095015936
// MI455X (gfx1250) — compile-verified
//
#include <hip/hip_runtime.h>

// ---------------------------------------------------------------------------
// TF AdaptiveSoftmax forward on MI455X (gfx1250).
//   out [512 x 267735] f32 full log-probs  +  scalar mean NLL.
// Bandwidth-bound: 548 MB output write dominates (>23 us at 23.3 TB/s).
// All GEMMs via V_WMMA_F32_16X16X4_F32; logits are recomputed for the
// max / sum-exp / write passes instead of spilled (weights ~147 MB fit in the
// 192 MB L2, so passes 2-3 re-read from L2). HBM traffic ~ out + weights.
// Inner K-loop: branchless (OOB columns clamped to a valid W row, masked only
// in the epilogue) and split into a batched load phase (20 global_load_b64
// into distinct regs) + WMMA burst phase, so loads are not serialized against
// individual matrix ops.
// ---------------------------------------------------------------------------

typedef float v2f __attribute__((ext_vector_type(2)));
typedef float v8f __attribute__((ext_vector_type(8)));

#define ROWS    512          // SEQ*BSZ tokens
#define VOCAB_N 267735

enum { M_STORE = 0, M_MAX = 1, M_SUM = 2, M_WRITE = 3 };

// order-monotonic encoding of f32 so atomicMax(uint) == float max
__device__ __forceinline__ unsigned enc_f32(float f) {
  unsigned u = __float_as_uint(f);
  return (u & 0x80000000u) ? ~u : (u | 0x80000000u);
}
__device__ __forceinline__ float dec_f32(unsigned u) {
  unsigned v = (u & 0x80000000u) ? (u & 0x7FFFFFFFu) : ~u;
  return __uint_as_float(v);
}

// One wave computes NT adjacent 16x16 f32 tiles (same 16 rows) of
//   C = X[512xK] * W[NxK]^T      via v_wmma_f32_16x16x4_f32.
// C/D layout: VGPR j, lanes 0-15: (row r0+j,   col c0+lane)
//             VGPR j, lanes16-31: (row r0+j+8, col c0+lane-16)
// A 16x4 f32: lane%16 = row M; lanes 0-15 hold K={k0,k0+1}, 16-31 K={k0+2,k0+3}
// B 4x16 f32: same pattern with lane%16 = col N (B fed from W rows).
template <int MODE, int NT>
__global__ void __launch_bounds__(32)
gemm_tile(const float* __restrict__ X, int ldx,
          const float* __restrict__ W,     // [N x K], row-major
          const float* __restrict__ bias,  // [N] or nullptr
          int N, int K,
          float* __restrict__ out, int ldo, long colBase,
          unsigned* __restrict__ rowMaxEnc,
          float* __restrict__ sumExp,
          const float* __restrict__ clusterLP, int clusterIdx)
{
  const int lane = threadIdx.x & 31;
  const int half = lane >> 4;          // 0 or 1
  const int l16  = lane & 15;
  const int r0   = blockIdx.y << 4;
  const int c0   = blockIdx.x * (16 * NT);

  int  colIdx[NT];
  bool colOK[NT];
  const float2* Wrow[NT];
#pragma unroll
  for (int t = 0; t < NT; ++t) {
    colIdx[t] = c0 + t * 16 + l16;
    colOK[t]  = (colIdx[t] < N);
    const int cc = colOK[t] ? colIdx[t] : (N - 1);   // clamp: always valid row
    Wrow[t] = (const float2*)(W + (size_t)cc * K);
  }

  v8f acc[NT];
#pragma unroll
  for (int t = 0; t < NT; ++t)
    acc[t] = (v8f){0.f, 0.f, 0.f, 0.f, 0.f, 0.f, 0.f, 0.f};

  const float2* Xrow = (const float2*)(X + (size_t)(r0 + l16) * ldx);

  // K is always a multiple of 16 here (1024/256/64/16).
  // Phase 1: batch 4 A-fragments + 4*NT B-fragments into distinct registers.
  // Phase 2: 4*NT WMMAs. Distinct destinations let the scheduler issue all
  // loads before the first wait instead of load->wait->wmma lockstep.
  for (int k0 = 0; k0 < K; k0 += 16) {
    float2 av[4];
    float2 bv[4][NT];
#pragma unroll
    for (int u = 0; u < 4; ++u) {
      const int kp = ((k0 + 4 * u) >> 1) + half;   // float2 idx: k + 2*half
      av[u] = Xrow[kp];
#pragma unroll
      for (int t = 0; t < NT; ++t) bv[u][t] = Wrow[t][kp];
    }
#pragma unroll
    for (int u = 0; u < 4; ++u) {
      v2f a; a.x = av[u].x; a.y = av[u].y;
#pragma unroll
      for (int t = 0; t < NT; ++t) {
        v2f b; b.x = bv[u][t].x; b.y = bv[u][t].y;
        acc[t] = __builtin_amdgcn_wmma_f32_16x16x4_f32(false, a, false, b,
                                                       (short)0, acc[t],
                                                       false, false);
      }
    }
  }

  if (MODE == M_STORE) {
#pragma unroll
    for (int t = 0; t < NT; ++t)
#pragma unroll
      for (int j = 0; j < 8; ++j) {
        const int row = r0 + j + (half << 3);
        if (colOK[t]) out[(size_t)row * ldo + colIdx[t]] = acc[t][j];
      }
    return;
  }

  float bb[NT];
#pragma unroll
  for (int t = 0; t < NT; ++t)
    bb[t] = (colOK[t] && bias) ? bias[colIdx[t]] : 0.f;

  if (MODE == M_MAX) {
#pragma unroll
    for (int j = 0; j < 8; ++j) {
      float v = -__builtin_inff();
#pragma unroll
      for (int t = 0; t < NT; ++t)
        if (colOK[t]) v = fmaxf(v, acc[t][j] + bb[t]);
      v = fmaxf(v, __shfl_xor(v, 1, 32));
      v = fmaxf(v, __shfl_xor(v, 2, 32));
      v = fmaxf(v, __shfl_xor(v, 4, 32));
      v = fmaxf(v, __shfl_xor(v, 8, 32));
      if (l16 == 0) {
        const int row = r0 + j + (half << 3);
        atomicMax(&rowMaxEnc[row], enc_f32(v));
      }
    }
  } else if (MODE == M_SUM) {
#pragma unroll
    for (int j = 0; j < 8; ++j) {
      const int row = r0 + j + (half << 3);
      const float mrow = dec_f32(rowMaxEnc[row]);
      float v = 0.f;
#pragma unroll
      for (int t = 0; t < NT; ++t)
        if (colOK[t]) v += __expf(acc[t][j] + bb[t] - mrow);
      v += __shfl_xor(v, 1, 32);
      v += __shfl_xor(v, 2, 32);
      v += __shfl_xor(v, 4, 32);
      v += __shfl_xor(v, 8, 32);
      if (l16 == 0) atomicAdd(&sumExp[row], v);
    }
  } else { // M_WRITE
#pragma unroll
    for (int j = 0; j < 8; ++j) {
      const int row  = r0 + j + (half << 3);
      const float lz = dec_f32(rowMaxEnc[row]) + __logf(sumExp[row]);
      const float o  = clusterLP ? clusterLP[row * 3 + clusterIdx] : 0.f;
#pragma unroll
      for (int t = 0; t < NT; ++t)
        if (colOK[t])
          out[(size_t)row * VOCAB_N + colBase + colIdx[t]] =
              acc[t][j] + bb[t] - lz + o;
    }
  }
}

// 3 cluster-pseudo-token head logits per row (dot over K=1024), merged into
// the head row max.  One wave per row.
__global__ void __launch_bounds__(32)
cluster_merge(const float* __restrict__ Y0, const float* __restrict__ cw,
              const float* __restrict__ cb,
              float* __restrict__ clusterLogit, unsigned* __restrict__ rowMaxEnc0)
{
  const int row  = blockIdx.x;
  const int lane = threadIdx.x;
  const float* y = Y0 + (size_t)row * 1024;
  for (int j = 0; j < 3; ++j) {
    float s = 0.f;
    for (int d = lane; d < 1024; d += 32) s += y[d] * cw[j * 1024 + d];
    s += __shfl_xor(s, 1, 32);
    s += __shfl_xor(s, 2, 32);
    s += __shfl_xor(s, 4, 32);
    s += __shfl_xor(s, 8, 32);
    s += __shfl_xor(s, 16, 32);
    if (lane == 0) {
      const float lg = s + cb[j];
      clusterLogit[row * 3 + j] = lg;
      atomicMax(&rowMaxEnc0[row], enc_f32(lg));
    }
  }
}

// Fold cluster exp-terms into head sum; produce clusterLP = head log-probs of
// the 3 pseudo tokens (offsets for the tail clusters).
__global__ void finalize_head(const float* __restrict__ clusterLogit,
                              const unsigned* __restrict__ rowMaxEnc0,
                              float* __restrict__ sumExp0,
                              float* __restrict__ clusterLP)
{
  const int row = blockIdx.x * blockDim.x + threadIdx.x;
  if (row >= ROWS) return;
  const float mx = dec_f32(rowMaxEnc0[row]);
  float S = sumExp0[row];
  for (int j = 0; j < 3; ++j) S += __expf(clusterLogit[row * 3 + j] - mx);
  sumExp0[row] = S;
  const float logZ = mx + __logf(S);
  for (int j = 0; j < 3; ++j)
    clusterLP[row * 3 + j] = clusterLogit[row * 3 + j] - logZ;
}

__global__ void init_stats(unsigned* __restrict__ p, int n)
{
  const int i = blockIdx.x * blockDim.x + threadIdx.x;
  if (i < n) p[i] = 0u;   // 0u == -inf in enc order; 0u bits == 0.0f for sums
}

// loss = -mean_t out[t, target[t]] ; single 512-thread block.
__global__ void loss_kernel(const int* __restrict__ target,
                            const float* __restrict__ out,
                            float* __restrict__ d_loss)
{
  __shared__ float sh[ROWS];
  const int t = threadIdx.x;
  sh[t] = out[(size_t)t * VOCAB_N + target[t]];
  __syncthreads();
  for (int s = ROWS / 2; s > 0; s >>= 1) {
    if (t < s) sh[t] += sh[t + s];
    __syncthreads();
  }
  if (t == 0) d_loss[0] = -sh[0] / (float)ROWS;
}

extern "C" void kernel_launch(void* const* d_in, const int* in_sizes, int n_in,
                              void* d_out, int out_size, void* d_ws, size_t ws_size,
                              hipStream_t stream)
{
  const float* hidden  = (const float*)d_in[0];
  const int*   target  = (const int*)d_in[1];
  const float* cw      = (const float*)d_in[2];
  const float* cb      = (const float*)d_in[3];
  const float* proj[4] = {(const float*)d_in[4], (const float*)d_in[7],
                          (const float*)d_in[10], (const float*)d_in[13]};
  const float* W[4]    = {(const float*)d_in[5], (const float*)d_in[8],
                          (const float*)d_in[11], (const float*)d_in[14]};
  const float* B[4]    = {(const float*)d_in[6], (const float*)d_in[9],
                          (const float*)d_in[12], (const float*)d_in[15]};
  float* out = (float*)d_out;

  const int  K[4]    = {1024, 256, 64, 16};          // d_embed per cluster
  const int  N[4]    = {20000, 20000, 160000, 67735};
  const long base[4] = {0, 20000, 40000, 200000};    // CUT_ENDS

  constexpr int NT = 4;                              // 64 columns per wave

  // workspace layout (~2.8 MB, all f32/u32)
  float* ws = (float*)d_ws;
  size_t off = 0;
  float* Y[4];
  for (int i = 0; i < 4; ++i) { Y[i] = ws + off; off += (size_t)ROWS * K[i]; }
  unsigned* rowMaxEnc = (unsigned*)(ws + off); off += 4 * ROWS;
  float* sumExp  = ws + off; off += 4 * ROWS;
  float* clLogit = ws + off; off += 3 * ROWS;
  float* clLP    = ws + off; off += 3 * ROWS;
  const int statDwords = (4 + 4 + 3 + 3) * ROWS;     // 7168

  init_stats<<<(statDwords + 255) / 256, 256, 0, stream>>>(rowMaxEnc, statDwords);

  // 1) projections  Y_i = hidden * proj_i^T   (N = K[i], K = 1024)
  for (int i = 0; i < 4; ++i) {
    const int ct = (K[i] + 16 * NT - 1) / (16 * NT);
    gemm_tile<M_STORE, NT><<<dim3(ct, ROWS / 16), 32, 0, stream>>>(
        hidden, 1024, proj[i], nullptr, K[i], 1024,
        Y[i], K[i], 0, nullptr, nullptr, nullptr, 0);
  }

  // 2) per-row max of logits (atomic, order-encoded)
  for (int i = 0; i < 4; ++i) {
    const int ct = (N[i] + 16 * NT - 1) / (16 * NT);
    gemm_tile<M_MAX, NT><<<dim3(ct, ROWS / 16), 32, 0, stream>>>(
        Y[i], K[i], W[i], B[i], N[i], K[i],
        nullptr, 0, 0, rowMaxEnc + i * ROWS, nullptr, nullptr, 0);
  }
  cluster_merge<<<ROWS, 32, 0, stream>>>(Y[0], cw, cb, clLogit, rowMaxEnc);

  // 3) per-row sum of exp (recompute GEMM; weights hit L2)
  for (int i = 0; i < 4; ++i) {
    const int ct = (N[i] + 16 * NT - 1) / (16 * NT);
    gemm_tile<M_SUM, NT><<<dim3(ct, ROWS / 16), 32, 0, stream>>>(
        Y[i], K[i], W[i], B[i], N[i], K[i],
        nullptr, 0, 0, rowMaxEnc + i * ROWS, sumExp + i * ROWS, nullptr, 0);
  }
  finalize_head<<<2, 256, 0, stream>>>(clLogit, rowMaxEnc, sumExp, clLP);

  // 4) final write pass: log-probs straight to d_out (548 MB stream-out)
  for (int i = 0; i < 4; ++i) {
    const int ct = (N[i] + 16 * NT - 1) / (16 * NT);
    gemm_tile<M_WRITE, NT><<<dim3(ct, ROWS / 16), 32, 0, stream>>>(
        Y[i], K[i], W[i], B[i], N[i], K[i],
        out, 0, base[i], rowMaxEnc + i * ROWS, sumExp + i * ROWS,
        (i == 0 ? nullptr : clLP), i - 1);
  }

  // 5) mean NLL at d_out[512*VOCAB]
  loss_kernel<<<1, ROWS, 0, stream>>>(target, out, out + (size_t)ROWS * VOCAB_N);
}